// EdgeConditionedConv_77979426226293
// MI455X (gfx1250) — compile-verified
//
#include <hip/hip_runtime.h>
#include <hip/hip_bf16.h>

// Problem constants (from reference setup_inputs)
#define BB_ 8     // batch
#define NN_ 128   // nodes
#define FF_ 32    // in features
#define SS_ 8     // edge features
#define HH_ 32    // hidden width
#define CC_ 32    // out channels (F_)

typedef float v2f __attribute__((ext_vector_type(2)));
typedef float v8f __attribute__((ext_vector_type(8)));

// ---------------------------------------------------------------------------
// Kernel 1: rowinv[a,b] = 1/max(sum_i fltr[a,b,i],1e-11)
//           bb[a,i,c]   = sum_f nf[a,i,f] * b_out[c*F+f]
// 1024 blocks x 32 threads (one wave per (a,row))
// ---------------------------------------------------------------------------
__global__ void ecc_prep_kernel(const float* __restrict__ nf,
                                const float* __restrict__ fltr,
                                const float* __restrict__ b_out,
                                float* __restrict__ rowinv,
                                float* __restrict__ bb) {
    const int blk  = blockIdx.x;
    const int a    = blk >> 7;
    const int r    = blk & (NN_ - 1);
    const int lane = threadIdx.x;

    // row sum of the filter (adjacency) row
    const float* frow = fltr + ((size_t)a * NN_ + r) * NN_;
    float s = frow[lane] + frow[lane + 32] + frow[lane + 64] + frow[lane + 96];
#pragma unroll
    for (int off = 16; off > 0; off >>= 1) s += __shfl_xor(s, off, 32);
    if (lane == 0) rowinv[a * NN_ + r] = 1.0f / fmaxf(s, 1e-11f);

    // bb: lane = output channel c
    const float* nrow = nf + ((size_t)a * NN_ + r) * FF_;
    const float* brow = b_out + (size_t)lane * FF_;
    float acc = 0.0f;
#pragma unroll
    for (int f = 0; f < FF_; ++f) acc = fmaf(nrow[f], brow[f], acc);
    bb[((size_t)a * NN_ + r) * CC_ + lane] = acc;
}

// ---------------------------------------------------------------------------
// Kernel 2: M[a,i,h,c] = sum_f nf[a,i,f] * W_out[h*1024 + c*32 + f]
// 1024 blocks (one per (a,i)) x 256 threads; 67 MFLOP total.
// ---------------------------------------------------------------------------
__global__ void ecc_m_kernel(const float* __restrict__ nf,
                             const float* __restrict__ W_out,
                             float* __restrict__ M) {
    const int blk = blockIdx.x;  // a*N + i
    __shared__ float snf[FF_];
    if (threadIdx.x < FF_) snf[threadIdx.x] = nf[(size_t)blk * FF_ + threadIdx.x];
    __syncthreads();

    for (int j = threadIdx.x; j < HH_ * CC_; j += 256) {
        const int h = j >> 5;
        const int c = j & 31;
        const float* wrow = W_out + (size_t)h * (CC_ * FF_) + c * FF_;
        float acc = 0.0f;
#pragma unroll
        for (int f = 0; f < FF_; ++f) acc = fmaf(snf[f], wrow[f], acc);
        M[(size_t)blk * (HH_ * CC_) + j] = acc;   // layout [a,i,h,c]
    }
}

// ---------------------------------------------------------------------------
// Kernel 3 (main): per (batch, 16-row tile) block, 4 waves.
// Wave w handles i in [32w, 32w+32). Per i: compute the 32 h-values
// (relu(ef@W0+b0) * fltr_n) on the VALU, then 8 (+1 for bb) k-steps of
// v_wmma_f32_16x16x4_f32 against M[a,i,:,:] for two 16-col C tiles.
// Wave partials reduced through LDS, bias added, result stored.
// ---------------------------------------------------------------------------
__global__ void __launch_bounds__(128)
ecc_main_kernel(const float* __restrict__ fltr,
                const float* __restrict__ ef,
                const float* __restrict__ W0,
                const float* __restrict__ b0,
                const float* __restrict__ bias,
                const float* __restrict__ rowinv,
                const float* __restrict__ bbuf,
                const float* __restrict__ M,
                float* __restrict__ out) {
    const int a  = blockIdx.x >> 3;         // batch
    const int bt = blockIdx.x & 7;          // 16-row tile of b
    const int w    = threadIdx.x >> 5;      // wave id 0..3 (K split over i)
    const int lane = threadIdx.x & 31;
    const int lo   = lane & 15;             // row within tile / col within tile
    const int hi   = lane >> 4;             // half-wave select

    __shared__ float sW0[SS_ * HH_];        // 256
    __shared__ float sb0[HH_];              // 32
    __shared__ float red[4][16][CC_];       // 8 KB wave partials

    for (int idx = threadIdx.x; idx < SS_ * HH_ + HH_; idx += 128) {
        if (idx < SS_ * HH_) sW0[idx] = W0[idx];
        else                 sb0[idx - SS_ * HH_] = b0[idx - SS_ * HH_];
    }
    __syncthreads();

    const int b = bt * 16 + lo;             // global output row (node)
    const float rinv = rowinv[a * NN_ + b];

    v8f acc0 = {0.f,0.f,0.f,0.f,0.f,0.f,0.f,0.f};   // cols 0..15
    v8f acc1 = {0.f,0.f,0.f,0.f,0.f,0.f,0.f,0.f};   // cols 16..31

    const int i0 = w * (NN_ / 4);
    for (int ii = 0; ii < NN_ / 4; ++ii) {
        const int i = i0 + ii;
        const float fn = fltr[((size_t)a * NN_ + b) * NN_ + i] * rinv;

        // edge feature vector for (a,b,i)
        const float* ep = ef + (((size_t)a * NN_ + b) * NN_ + i) * SS_;
        float e[SS_];
#pragma unroll
        for (int s = 0; s < SS_; ++s) e[s] = ep[s];

        // 16 of the 32 h-values this lane contributes (paired halves cover all 32):
        // j -> h = 4*(j>>1) + 2*hi + (j&1)
        float hv[16];
#pragma unroll
        for (int j = 0; j < 16; ++j) {
            const int hidx = 4 * (j >> 1) + 2 * hi + (j & 1);
            float acc = sb0[hidx];
#pragma unroll
            for (int s = 0; s < SS_; ++s) acc = fmaf(e[s], sW0[s * HH_ + hidx], acc);
            hv[j] = fmaxf(acc, 0.0f) * fn;
        }

        // 8 WMMA k-steps over h (K=4 each), two column tiles
        const float* mbase = M + ((size_t)a * NN_ + i) * (HH_ * CC_);
#pragma unroll
        for (int t = 0; t < 8; ++t) {
            v2f af;  af.x = hv[2 * t]; af.y = hv[2 * t + 1];   // A[row, 4t+2hi+{0,1}]
            const float* mb = mbase + (4 * t + 2 * hi) * CC_;
            v2f bf0; bf0.x = mb[lo];        bf0.y = mb[CC_ + lo];
            v2f bf1; bf1.x = mb[16 + lo];   bf1.y = mb[CC_ + 16 + lo];
            acc0 = __builtin_amdgcn_wmma_f32_16x16x4_f32(false, af, false, bf0,
                                                         (short)0, acc0, false, false);
            acc1 = __builtin_amdgcn_wmma_f32_16x16x4_f32(false, af, false, bf1,
                                                         (short)0, acc1, false, false);
        }

        // extra k-step folding in the b_out term: + fn * bb[a,i,c]
        {
            const float* bbp = bbuf + ((size_t)a * NN_ + i) * CC_;
            const float t0 = bbp[lo];
            const float t1 = bbp[16 + lo];
            v2f af;  af.x = (hi == 0) ? fn : 0.0f; af.y = 0.0f;
            v2f bf0; bf0.x = (hi == 0) ? t0 : 0.0f; bf0.y = 0.0f;
            v2f bf1; bf1.x = (hi == 0) ? t1 : 0.0f; bf1.y = 0.0f;
            acc0 = __builtin_amdgcn_wmma_f32_16x16x4_f32(false, af, false, bf0,
                                                         (short)0, acc0, false, false);
            acc1 = __builtin_amdgcn_wmma_f32_16x16x4_f32(false, af, false, bf1,
                                                         (short)0, acc1, false, false);
        }
    }

    // C/D layout: lane holds C[row = v + 8*hi, col = lo (tile0) / 16+lo (tile1)]
#pragma unroll
    for (int v = 0; v < 8; ++v) {
        red[w][hi * 8 + v][lo]      = acc0[v];
        red[w][hi * 8 + v][16 + lo] = acc1[v];
    }
    __syncthreads();

    // reduce the 4 wave partials, add bias, store
#pragma unroll
    for (int k = 0; k < 4; ++k) {
        const int idx = threadIdx.x + k * 128;   // 0..511
        const int row = idx >> 5;
        const int c   = idx & 31;
        const float sum = red[0][row][c] + red[1][row][c] +
                          red[2][row][c] + red[3][row][c] + bias[c];
        out[((size_t)a * NN_ + bt * 16 + row) * CC_ + c] = sum;
    }
}

// ---------------------------------------------------------------------------
extern "C" void kernel_launch(void* const* d_in, const int* in_sizes, int n_in,
                              void* d_out, int out_size, void* d_ws, size_t ws_size,
                              hipStream_t stream) {
    const float* nf    = (const float*)d_in[0];  // (8,128,32)
    const float* fltr  = (const float*)d_in[1];  // (8,128,128)
    const float* ef    = (const float*)d_in[2];  // (8,128,128,8)
    const float* W0    = (const float*)d_in[3];  // (8,32)
    const float* b0    = (const float*)d_in[4];  // (32,)
    const float* W_out = (const float*)d_in[5];  // (32,1024)
    const float* b_out = (const float*)d_in[6];  // (1024,)
    const float* bias  = (const float*)d_in[7];  // (32,)
    float* out = (float*)d_out;                  // (8,128,32)

    // workspace layout (floats): rowinv[1024] | bb[32768] | M[1048576]  (~4.3 MB)
    float* rowinv = (float*)d_ws;
    float* bb     = rowinv + BB_ * NN_;
    float* M      = bb + (size_t)BB_ * NN_ * CC_;

    ecc_prep_kernel<<<BB_ * NN_, 32, 0, stream>>>(nf, fltr, b_out, rowinv, bb);
    ecc_m_kernel<<<BB_ * NN_, 256, 0, stream>>>(nf, W_out, M);
    ecc_main_kernel<<<BB_ * (NN_ / 16), 128, 0, stream>>>(
        fltr, ef, W0, b0, bias, rowinv, bb, M, out);
}